// LinearAttentionHead_37512244363842
// MI455X (gfx1250) — compile-verified
//
#include <hip/hip_runtime.h>
#include <hip/hip_bf16.h>

// Linformer attention for MI455X (gfx1250), wave32, FP32 WMMA path.
// Shapes: B=16, S=4096, D=64, KP=256.
//
//   Kp[b,d,k]  = sum_s K[b,s,d]*E_w[k,s] + E_b[k]
//   Vp[b,k,d]  = sum_s V[b,s,d]*F_w[k,s] + F_b[k]
//   scores     = (Q @ Kp) * 1/sqrt(D);  attn = softmax(scores)
//   out        = attn @ Vp
//
// Kp/Vp are stored in scratch PRE-PACKED in WMMA B-fragment order so each
// lane's B-fragment (rows k,k+1 at column n) is one aligned float2:
//   Kpp[b][d>>1][kp][d&1]   (for scores GEMM, K-dim = d)
//   Vpp[b][kp>>1][d][kp&1]  (for output GEMM, K-dim = kp)
// Per-tile fragment preloading + independent accumulator chains let loads
// overlap the WMMA bursts instead of the load->wait->wmma lockstep of r1.

typedef float v2f __attribute__((ext_vector_type(2)));
typedef float v8f __attribute__((ext_vector_type(8)));

#define LIN_B  16
#define LIN_S  4096
#define LIN_D  64
#define LIN_KP 256

__device__ __forceinline__ v8f wmma_f32(v2f a, v2f b, v8f c) {
    return __builtin_amdgcn_wmma_f32_16x16x4_f32(false, a, false, b, (short)0, c,
                                                 false, false);
}

// ---------------------------------------------------------------------------
// Projection: acc(KPxD tile) = W(KPxS) * X[b](SxD) + bias, one tile per wave.
// A(m=kp,k=s) = W[kp][s], B(k=s,n=d) = X[b][s][d].
// KT_PACK=true  -> store B-fragment-packed over d  (Kpp, feeds scores GEMM)
// KT_PACK=false -> store B-fragment-packed over kp (Vpp, feeds output GEMM)
// ---------------------------------------------------------------------------
template <bool KT_PACK>
__global__ __launch_bounds__(32) void lin_proj_kernel(
    const float* __restrict__ X, const float* __restrict__ W,
    const float* __restrict__ bias, float* __restrict__ outP) {
    const int S = LIN_S, D = LIN_D, KP = LIN_KP;

    const int blk  = blockIdx.x;
    const int b    = blk >> 6;     // 64 tiles per batch
    const int tile = blk & 63;
    const int mt   = tile >> 2;    // kp tile 0..15
    const int nt   = tile & 3;     // d  tile 0..3
    const int l    = threadIdx.x;
    const int lh   = l >> 4;
    const int ll   = l & 15;
    const int kp0  = mt * 16;
    const int d0   = nt * 16;

    const float* wrow = W + (size_t)(kp0 + ll) * S + lh * 2;  // A frags
    const float* xcol = X + (size_t)b * S * D + d0 + ll;      // B frags

    v8f acc = {};
    for (int s0 = 0; s0 < S; s0 += 32) {  // 8 k-steps per pass
        v2f a[8], bf[8];
#pragma unroll
        for (int j = 0; j < 8; ++j)
            a[j] = *(const v2f*)(wrow + s0 + j * 4);
#pragma unroll
        for (int j = 0; j < 8; ++j) {
            const float* xp = xcol + (size_t)(s0 + j * 4 + lh * 2) * D;
            bf[j].x = xp[0];
            bf[j].y = xp[D];
        }
#pragma unroll
        for (int j = 0; j < 8; ++j) acc = wmma_f32(a[j], bf[j], acc);
    }

    // C/D(m,n): vgpr v -> m = kp0 + v + 8*lh, n = d0 + ll. Bias per kp row.
#pragma unroll
    for (int v = 0; v < 8; ++v) acc[v] += bias[kp0 + 8 * lh + v];

    const int d = d0 + ll;
    if (KT_PACK) {
        // Kpp idx(d,kp) = (d>>1)*2*KP + kp*2 + (d&1)
        float* p = outP + (size_t)b * D * KP + (size_t)(d >> 1) * (2 * KP) +
                   (size_t)(kp0 + 8 * lh) * 2 + (d & 1);
#pragma unroll
        for (int v = 0; v < 8; ++v) p[2 * v] = acc[v];
    } else {
        // Vpp idx(kp,d) = (kp>>1)*2*D + d*2 + (kp&1)
        float* base = outP + (size_t)b * KP * D + d * 2;
#pragma unroll
        for (int v = 0; v < 8; ++v) {
            const int kp = kp0 + 8 * lh + v;
            base[(size_t)(kp >> 1) * (2 * D) + (kp & 1)] = acc[v];
        }
    }
}

// ---------------------------------------------------------------------------
// Fused scores + softmax + output GEMM. One wave per (batch, 16-row s-strip).
// ---------------------------------------------------------------------------
#define LDS_STRIDE 260  // 260 % 64 == 4 -> conflict-free for our two patterns

__global__ __launch_bounds__(32) void lin_attn_kernel(
    const float* __restrict__ Q, const float* __restrict__ Kpp,
    const float* __restrict__ Vpp, float* __restrict__ out,
    float* __restrict__ attn) {
    const int S = LIN_S, D = LIN_D, KP = LIN_KP;
    const float scale = 0.125f;  // 1/sqrt(64)

    __shared__ float lds[16 * LDS_STRIDE];

    const int blk = blockIdx.x;
    const int b   = blk >> 8;  // 256 strips per batch
    const int s0  = (blk & 255) * 16;
    const int l   = threadIdx.x;
    const int lh  = l >> 4;
    const int ll  = l & 15;

    // --- Q A-fragments: A(m=s_local,k=d), 16 K-steps of 4 -------------------
    v2f qa[16];
    const float* qrow = Q + (size_t)b * S * D + (size_t)(s0 + ll) * D + lh * 2;
#pragma unroll
    for (int kk = 0; kk < 16; ++kk) qa[kk] = *(const v2f*)(qrow + kk * 4);

    // --- scores = Q @ Kp: preload 16 B-frags/tile, then 16 chained WMMAs ----
    v8f sc[16];
    const float* kpB = Kpp + (size_t)b * D * KP + ll * 2;
#pragma unroll
    for (int t = 0; t < 16; ++t) {
        v2f bf[16];
#pragma unroll
        for (int kk = 0; kk < 16; ++kk)  // one coalesced b64 per fragment
            bf[kk] = *(const v2f*)(kpB + (size_t)(kk * 2 + lh) * (2 * KP) + t * 32);
        v8f c = {};
#pragma unroll
        for (int kk = 0; kk < 16; ++kk) c = wmma_f32(qa[kk], bf[kk], c);
        sc[t] = c;
    }

    // --- softmax over 256 cols per row; row m = v + 8*lh spans 16 lanes -----
#pragma unroll
    for (int v = 0; v < 8; ++v) {
        float mx = sc[0][v];
#pragma unroll
        for (int t = 1; t < 16; ++t) mx = fmaxf(mx, sc[t][v]);
        mx = fmaxf(mx, __shfl_xor(mx, 1, 32));
        mx = fmaxf(mx, __shfl_xor(mx, 2, 32));
        mx = fmaxf(mx, __shfl_xor(mx, 4, 32));
        mx = fmaxf(mx, __shfl_xor(mx, 8, 32));  // stays within 16-lane halves
        float sum = 0.f;
#pragma unroll
        for (int t = 0; t < 16; ++t) {
            float p = __expf((sc[t][v] - mx) * scale);
            sc[t][v] = p;
            sum += p;
        }
        sum += __shfl_xor(sum, 1, 32);
        sum += __shfl_xor(sum, 2, 32);
        sum += __shfl_xor(sum, 4, 32);
        sum += __shfl_xor(sum, 8, 32);
        const float r = 1.0f / sum;
#pragma unroll
        for (int t = 0; t < 16; ++t) sc[t][v] *= r;
    }

    // --- store attn (2nd output) + stage into LDS (D-layout -> A-layout) ---
    float* arow = attn + (size_t)b * S * KP;
#pragma unroll
    for (int t = 0; t < 16; ++t) {
#pragma unroll
        for (int v = 0; v < 8; ++v) {
            const int m   = v + 8 * lh;   // strip row
            const int col = t * 16 + ll;  // kp
            arow[(size_t)(s0 + m) * KP + col] = sc[t][v];
            lds[m * LDS_STRIDE + col]         = sc[t][v];
        }
    }
    __syncthreads();

    // --- out = attn @ Vp: 4 independent chains share one LDS A-frag ---------
    const float* vpB = Vpp + (size_t)b * KP * D + ll * 2;
    v8f oc[4] = {};
#pragma unroll 4
    for (int kk = 0; kk < 64; ++kk) {
        v2f a = *(const v2f*)(&lds[ll * LDS_STRIDE + kk * 4 + lh * 2]);
        v2f bf[4];
#pragma unroll
        for (int td = 0; td < 4; ++td)  // one coalesced b64 per fragment
            bf[td] = *(const v2f*)(vpB + (size_t)(kk * 2 + lh) * (2 * D) + td * 32);
#pragma unroll
        for (int td = 0; td < 4; ++td) oc[td] = wmma_f32(a, bf[td], oc[td]);
    }
#pragma unroll
    for (int td = 0; td < 4; ++td) {
#pragma unroll
        for (int v = 0; v < 8; ++v)
            out[(size_t)b * S * D + (size_t)(s0 + v + 8 * lh) * D + td * 16 + ll] =
                oc[td][v];
    }
}

extern "C" void kernel_launch(void* const* d_in, const int* in_sizes, int n_in,
                              void* d_out, int out_size, void* d_ws, size_t ws_size,
                              hipStream_t stream) {
    (void)in_sizes; (void)n_in; (void)out_size; (void)ws_size;
    const float* Q   = (const float*)d_in[0];
    const float* K   = (const float*)d_in[1];
    const float* V   = (const float*)d_in[2];
    const float* E_w = (const float*)d_in[3];
    const float* E_b = (const float*)d_in[4];
    const float* F_w = (const float*)d_in[5];
    const float* F_b = (const float*)d_in[6];

    float* out  = (float*)d_out;                        // [B,S,D]
    float* attn = out + (size_t)LIN_B * LIN_S * LIN_D;  // [B,S,KP]

    float* Kpp = (float*)d_ws;                          // packed Kp, 1MB
    float* Vpp = Kpp + (size_t)LIN_B * LIN_D * LIN_KP;  // packed Vp, 1MB

    const int projBlocks = LIN_B * 16 * 4;        // 1024 tile-waves
    const int attnBlocks = LIN_B * (LIN_S / 16);  // 4096 strip-waves

    lin_proj_kernel<true ><<<projBlocks, 32, 0, stream>>>(K, E_w, E_b, Kpp);
    lin_proj_kernel<false><<<projBlocks, 32, 0, stream>>>(V, F_w, F_b, Vpp);
    lin_attn_kernel<<<attnBlocks, 32, 0, stream>>>(Q, Kpp, Vpp, out, attn);
}